// hybrid_block_85882166050972
// MI455X (gfx1250) — compile-verified
//
#include <hip/hip_runtime.h>

#define Bb 8
#define Cc 128
#define C2 256
#define Nn 2000
#define Ss 128
#define Hh 4
#define HD 32

typedef __attribute__((ext_vector_type(2))) float v2f;
typedef __attribute__((ext_vector_type(8))) float v8f;

__device__ __forceinline__ v8f wmma_f32_k4(v2f a, v2f b, v8f c) {
  // D = A(16x4) * B(4x16) + C, fp32 — V_WMMA_F32_16X16X4_F32
  return __builtin_amdgcn_wmma_f32_16x16x4_f32(false, a, false, b, (short)0, c,
                                               false, false);
}

// --- gfx1250 async global->LDS path (guarded; falls back to direct loads) ---
#if __has_builtin(__builtin_amdgcn_global_load_async_to_lds_b128)
#define HAS_ASYNC_LDS 1
typedef int v4i __attribute__((vector_size(16)));
typedef __attribute__((address_space(1))) v4i v4i_gl;  // global (__device__)
typedef __attribute__((address_space(3))) v4i v4i_sh;  // LDS (__shared__)
#if __has_builtin(__builtin_amdgcn_s_wait_asynccnt)
#define WAIT_ASYNC0() __builtin_amdgcn_s_wait_asynccnt(0)
#define WAIT_ASYNC4() __builtin_amdgcn_s_wait_asynccnt(4)
#else
#define WAIT_ASYNC0() asm volatile("s_wait_asynccnt 0" ::: "memory")
#define WAIT_ASYNC4() asm volatile("s_wait_asynccnt 4" ::: "memory")
#endif
#else
#define HAS_ASYNC_LDS 0
#endif

// ---------------------------------------------------------------- gather ----
__global__ void k_gather(const float* __restrict__ src, const int* __restrict__ idx,
                         float* __restrict__ dst) {
  int t = blockIdx.x * blockDim.x + threadIdx.x;
  if (t >= Bb * Cc * Ss) return;
  int s = t % Ss; int c = (t / Ss) % Cc; int b = t / (Ss * Cc);
  int j = idx[b * Ss + s];
  dst[t] = src[((size_t)b * Cc + c) * Nn + j];
}

// ---------------------------------------------------- strided channel copy --
__global__ void k_copy_chan(const float* __restrict__ src, int sbs,
                            float* __restrict__ dst, int dbs, int Cn, int P) {
  int t = blockIdx.x * blockDim.x + threadIdx.x;
  if (t >= Bb * Cn * P) return;
  int p = t % P; int c = (t / P) % Cn; int b = t / (P * Cn);
  dst[(size_t)b * dbs + (size_t)c * P + p] = src[(size_t)b * sbs + (size_t)c * P + p];
}

// ------------------------------------------- 1x1 conv as GEMM via WMMA fp32 -
// Y[b,o,p] = sum_i W[o,i]*X[b,i,p] (+bias[o]) (+res[b,o,p])
// One wave per 16(o) x 16(p) tile. O,P multiples of 16; I multiple of 32.
// B-panel (32 K-rows x 16 points) is double-buffered through LDS with
// GLOBAL_LOAD_ASYNC_TO_LDS_B128 (ASYNCcnt) when the toolchain supports it.
__global__ void k_conv_wmma(const float* __restrict__ X, const float* __restrict__ W,
                            const float* __restrict__ bias, const float* __restrict__ res,
                            float* __restrict__ Y,
                            int O, int I, int P, int xbs, int ybs, int rbs, int wld) {
  int lane = threadIdx.x;
  int mrow = lane & 15, kh = lane >> 4;
  int p0 = blockIdx.x * 16;
  int o0 = blockIdx.y * 16;
  int b  = blockIdx.z;
  const float* Xb = X + (size_t)b * xbs;
  const float* Wr = W + (size_t)(o0 + mrow) * wld;
  v8f acc = {0.f, 0.f, 0.f, 0.f, 0.f, 0.f, 0.f, 0.f};
#if HAS_ASYNC_LDS
  __shared__ float lbx[2][32 * 16];
  int srow = lane >> 2;        // 0..7
  int scol = (lane & 3) * 4;   // 0,4,8,12  (16B chunks)
  const float* Xp = Xb + p0 + scol;
  // stage panel k0..k0+31 into lbx[bufi]
  auto stage = [&](int bufi, int k0) {
#pragma unroll
    for (int c = 0; c < 4; ++c) {
      int row = c * 8 + srow;
      __builtin_amdgcn_global_load_async_to_lds_b128(
          (v4i_gl*)(Xp + (size_t)(k0 + row) * P),
          (v4i_sh*)&lbx[bufi][row * 16 + scol], 0, 0);
    }
  };
  stage(0, 0);
  int buf = 0;
  for (int k0 = 0; k0 < I; k0 += 32) {
    if (k0 + 32 < I) { stage(buf ^ 1, k0 + 32); WAIT_ASYNC4(); }
    else             { WAIT_ASYNC0(); }
    asm volatile("" ::: "memory");
    const float* bufp = lbx[buf];
#pragma unroll
    for (int kk = 0; kk < 32; kk += 4) {
      v2f a = *(const v2f*)(Wr + k0 + kk + 2 * kh);
      v2f bv;
      bv.x = bufp[(kk + 2 * kh) * 16 + mrow];
      bv.y = bufp[(kk + 2 * kh + 1) * 16 + mrow];
      acc = wmma_f32_k4(a, bv, acc);
    }
    asm volatile("" ::: "memory");
    buf ^= 1;
  }
#else
#pragma unroll 4
  for (int k0 = 0; k0 < I; k0 += 4) {
    v2f a = *(const v2f*)(Wr + k0 + 2 * kh);          // A: K=2*kh, 2*kh+1
    v2f bv;
    bv.x = Xb[(size_t)(k0 + 2 * kh) * P + p0 + mrow]; // B: K=2*kh
    bv.y = Xb[(size_t)(k0 + 2 * kh + 1) * P + p0 + mrow];
    acc = wmma_f32_k4(a, bv, acc);
  }
#endif
#pragma unroll
  for (int r = 0; r < 8; ++r) {
    int o = o0 + r + 8 * kh;
    int p = p0 + mrow;
    float v = acc[r];
    if (bias) v += bias[o];
    if (res)  v += res[(size_t)b * rbs + (size_t)o * P + p];
    Y[(size_t)b * ybs + (size_t)o * P + p] = v;
  }
}

// -------------------------------------------- attention scores (WMMA fp32) --
// Att[b,h,n,m] = scale * sum_d Q[b,h*HD+d,n]*K[b,h*HD+d,m]
__global__ void k_attn_score(const float* __restrict__ Q, const float* __restrict__ Km,
                             float* __restrict__ Att, int P1, int P2, float scale) {
  int lane = threadIdx.x;
  int mrow = lane & 15, kh = lane >> 4;
  int m0 = blockIdx.x * 16, n0 = blockIdx.y * 16;
  int bh = blockIdx.z; int b = bh / Hh, h = bh % Hh;
  const float* Qb = Q + (size_t)b * Cc * P1 + (size_t)h * HD * P1;
  const float* Kb = Km + (size_t)b * Cc * P2 + (size_t)h * HD * P2;
  v8f acc = {0.f, 0.f, 0.f, 0.f, 0.f, 0.f, 0.f, 0.f};
#pragma unroll
  for (int k0 = 0; k0 < HD; k0 += 4) {
    v2f a, bv;
    a.x  = Qb[(size_t)(k0 + 2 * kh) * P1 + n0 + mrow];
    a.y  = Qb[(size_t)(k0 + 2 * kh + 1) * P1 + n0 + mrow];
    bv.x = Kb[(size_t)(k0 + 2 * kh) * P2 + m0 + mrow];
    bv.y = Kb[(size_t)(k0 + 2 * kh + 1) * P2 + m0 + mrow];
    acc = wmma_f32_k4(a, bv, acc);
  }
  float* Ab = Att + (size_t)bh * P1 * P2;
#pragma unroll
  for (int r = 0; r < 8; ++r) {
    int n = n0 + r + 8 * kh;
    Ab[(size_t)n * P2 + m0 + mrow] = acc[r] * scale;
  }
}

// -------------------------------------------- attention apply (WMMA fp32) ---
// Out[b,h*HD+d,n] = sum_m V[b,h*HD+d,m]*Att[b,h,n,m]
__global__ void k_attn_apply(const float* __restrict__ Att, const float* __restrict__ V,
                             float* __restrict__ Out, int P1, int P2) {
  int lane = threadIdx.x;
  int mrow = lane & 15, kh = lane >> 4;
  int n0 = blockIdx.x * 16;
  int h  = blockIdx.y >> 1;
  int d0 = (blockIdx.y & 1) * 16;
  int b  = blockIdx.z;
  const float* Vb = V + (size_t)b * Cc * P2 + (size_t)(h * HD + d0) * P2;
  const float* Ab = Att + (size_t)(b * Hh + h) * P1 * P2;
  v8f acc = {0.f, 0.f, 0.f, 0.f, 0.f, 0.f, 0.f, 0.f};
  for (int m0 = 0; m0 < P2; m0 += 4) {
    v2f a  = *(const v2f*)(Vb + (size_t)mrow * P2 + m0 + 2 * kh);
    v2f bv = *(const v2f*)(Ab + (size_t)(n0 + mrow) * P2 + m0 + 2 * kh);
    acc = wmma_f32_k4(a, bv, acc);
  }
#pragma unroll
  for (int r = 0; r < 8; ++r) {
    int d = d0 + r + 8 * kh;
    Out[(size_t)b * Cc * P1 + (size_t)(h * HD + d) * P1 + n0 + mrow] = acc[r];
  }
}

// -------------------------------------------- BN1d stats (per-channel) ------
__global__ void k_bnstats(const float* __restrict__ X, int P, int xbs,
                          float* __restrict__ mean, float* __restrict__ istd, float eps) {
  __shared__ float s1[256], s2[256];
  int c = blockIdx.x;
  float a = 0.f, q = 0.f;
  for (int i = threadIdx.x; i < Bb * P; i += blockDim.x) {
    int b = i / P, p = i % P;
    float v = X[(size_t)b * xbs + (size_t)c * P + p];
    a += v; q += v * v;
  }
  s1[threadIdx.x] = a; s2[threadIdx.x] = q; __syncthreads();
  for (int st = blockDim.x >> 1; st > 0; st >>= 1) {
    if ((int)threadIdx.x < st) { s1[threadIdx.x] += s1[threadIdx.x + st]; s2[threadIdx.x] += s2[threadIdx.x + st]; }
    __syncthreads();
  }
  if (threadIdx.x == 0) {
    float m = s1[0] / (float)(Bb * P);
    float var = s2[0] / (float)(Bb * P) - m * m;
    mean[c] = m; istd[c] = rsqrtf(var + eps);
  }
}

__global__ void k_bnrelu(float* __restrict__ X, int Cn, int P, int xbs,
                         const float* __restrict__ mean, const float* __restrict__ istd,
                         const float* __restrict__ g, const float* __restrict__ be) {
  int t = blockIdx.x * blockDim.x + threadIdx.x;
  if (t >= Bb * Cn * P) return;
  int p = t % P; int c = (t / P) % Cn; int b = t / (P * Cn);
  size_t off = (size_t)b * xbs + (size_t)c * P + p;
  float v = (X[off] - mean[c]) * istd[c] * g[c] + be[c];
  X[off] = fmaxf(v, 0.f);
}

// ---------------------------------------------------- row softmax (in place)
__global__ void k_softmax(float* __restrict__ X, int L) {
  __shared__ float red[256];
  float* row = X + (size_t)blockIdx.x * L;
  float mx = -3.402823466e38f;
  for (int i = threadIdx.x; i < L; i += blockDim.x) mx = fmaxf(mx, row[i]);
  red[threadIdx.x] = mx; __syncthreads();
  for (int st = 128; st > 0; st >>= 1) {
    if ((int)threadIdx.x < st) red[threadIdx.x] = fmaxf(red[threadIdx.x], red[threadIdx.x + st]);
    __syncthreads();
  }
  mx = red[0]; __syncthreads();
  float sum = 0.f;
  for (int i = threadIdx.x; i < L; i += blockDim.x) { float e = expf(row[i] - mx); row[i] = e; sum += e; }
  red[threadIdx.x] = sum; __syncthreads();
  for (int st = 128; st > 0; st >>= 1) {
    if ((int)threadIdx.x < st) red[threadIdx.x] += red[threadIdx.x + st];
    __syncthreads();
  }
  float inv = 1.f / red[0];
  __syncthreads();
  for (int i = threadIdx.x; i < L; i += blockDim.x) row[i] *= inv;
}

// ---------------------------------------- per-(b,c) mean/var over P points --
__global__ void k_mv_bc(const float* __restrict__ X, int P, int xbs, int Cn,
                        float* __restrict__ mean, float* __restrict__ var) {
  __shared__ float s1[128], s2[128];
  int bc = blockIdx.x; int b = bc / Cn, c = bc % Cn;
  const float* row = X + (size_t)b * xbs + (size_t)c * P;
  float a = 0.f, q = 0.f;
  for (int i = threadIdx.x; i < P; i += blockDim.x) { float v = row[i]; a += v; q += v * v; }
  s1[threadIdx.x] = a; s2[threadIdx.x] = q; __syncthreads();
  for (int st = blockDim.x >> 1; st > 0; st >>= 1) {
    if ((int)threadIdx.x < st) { s1[threadIdx.x] += s1[threadIdx.x + st]; s2[threadIdx.x] += s2[threadIdx.x + st]; }
    __syncthreads();
  }
  if (threadIdx.x == 0) { float m = s1[0] / (float)P; mean[bc] = m; var[bc] = s2[0] / (float)P - m * m; }
}

// BN-over-IN closed form: vbar[c] = mean_b( var/(var+eps_in) )
__global__ void k_vbar(const float* __restrict__ var, float* __restrict__ vbar, int Cn, float epsin) {
  int c = blockIdx.x * blockDim.x + threadIdx.x;
  if (c >= Cn) return;
  float s = 0.f;
  for (int b = 0; b < Bb; ++b) { float v = var[b * Cn + c]; s += v / (v + epsin); }
  vbar[c] = s / (float)Bb;
}

// stage-1: Y = relu( (X - mu_bc) * rsqrt(var_bc+1e-3) * rsqrt(vbar_c+1e-5) * g[co+c] + be[co+c] )
__global__ void k_in_bn_relu(const float* __restrict__ X, float* __restrict__ Y,
                             const float* __restrict__ mean, const float* __restrict__ var,
                             const float* __restrict__ vbar, const float* __restrict__ g,
                             const float* __restrict__ be, int Cn, int P, int co) {
  int t = blockIdx.x * blockDim.x + threadIdx.x;
  if (t >= Bb * Cn * P) return;
  int c = (t / P) % Cn; int b = t / (P * Cn);
  int bc = b * Cn + c;
  float al = rsqrtf(var[bc] + 1e-3f) * rsqrtf(vbar[c] + 1e-5f) * g[co + c];
  float v = (X[t] - mean[bc]) * al + be[co + c];
  Y[t] = fmaxf(v, 0.f);
}

// stage-2 BN scale: s2c[o] = rsqrt( mean_b( (vq+vk)/(vq+vk+1e-3) ) + 1e-5 ) * g2[o]
__global__ void k_bn2scale(const float* __restrict__ vq, const float* __restrict__ vk,
                           const float* __restrict__ g2, float* __restrict__ s2c) {
  int o = blockIdx.x * blockDim.x + threadIdx.x;
  if (o >= C2) return;
  float acc = 0.f;
  for (int b = 0; b < Bb; ++b) { float v = vq[b * C2 + o] + vk[b * C2 + o]; acc += v / (v + 1e-3f); }
  s2c[o] = rsqrtf(acc / (float)Bb + 1e-5f) * g2[o];
}

// stage-2 apply (separable factor): Y = (A - mself)*rsqrt(vq+vk+1e-3)*s2c (+be2)
__global__ void k_stage2_apply(const float* __restrict__ A, const float* __restrict__ mself,
                               const float* __restrict__ vq, const float* __restrict__ vk,
                               const float* __restrict__ s2c, const float* __restrict__ be2,
                               float* __restrict__ Y) {
  int t = blockIdx.x * blockDim.x + threadIdx.x;
  if (t >= Bb * C2 * Ss) return;
  int o = (t / Ss) % C2; int b = t / (Ss * C2);
  int bo = b * C2 + o;
  float istd = rsqrtf(vq[bo] + vk[bo] + 1e-3f);
  float v = (A[t] - mself[bo]) * istd * s2c[o];
  if (be2) v += be2[o];
  Y[t] = v;
}

// out[b,p] = sum_c w[c]*X[b,c,p] + bias1 + bias2   (shortcut conv, O=1)
__global__ void k_dot1(const float* __restrict__ X, const float* __restrict__ w,
                       const float* __restrict__ b1, const float* __restrict__ b2,
                       float* __restrict__ outv, int Cn, int P, int xbs) {
  int t = blockIdx.x * blockDim.x + threadIdx.x;
  if (t >= Bb * P) return;
  int b = t / P, p = t % P;
  float acc = (b1 ? b1[0] : 0.f) + (b2 ? b2[0] : 0.f);
  const float* xb = X + (size_t)b * xbs + p;
  for (int c = 0; c < Cn; ++c) acc += w[c] * xb[(size_t)c * P];
  outv[t] = acc;
}

// s[b,n,m] = sum_o w2[o]*relu(Atq[b,o,n]+Atk[b,o,m]) + Sq[b,n] + Sk[b,m]
__global__ void k_sepsum(const float* __restrict__ Atq, const float* __restrict__ Atk,
                         const float* __restrict__ w2, const float* __restrict__ Sq,
                         const float* __restrict__ Sk, float* __restrict__ Sout) {
  __shared__ float aq[C2];
  __shared__ float ww[C2];
  int n = blockIdx.x, b = blockIdx.y, m = threadIdx.x;
  for (int o = threadIdx.x; o < C2; o += blockDim.x) {
    aq[o] = Atq[((size_t)b * C2 + o) * Ss + n];
    ww[o] = w2[o];
  }
  __syncthreads();
  float acc = Sq[b * Ss + n] + Sk[b * Ss + m];
  const float* Akb = Atk + (size_t)b * C2 * Ss + m;
#pragma unroll 4
  for (int o = 0; o < C2; ++o) acc += ww[o] * fmaxf(aq[o] + Akb[(size_t)o * Ss], 0.f);
  Sout[((size_t)b * Ss + n) * Ss + m] = acc;
}

// ======================================================== host orchestration
struct AP {
  const float *qb, *qw, *kb, *kw, *vb, *vw, *mhb, *mhw;
  const float *cat1b, *cat1w, *bnbe, *bng, *cat2b, *cat2w;
};

static AP ap_from(void* const* din, int base) {
  AP a;
  a.cat1b = (const float*)din[base + 0];  a.cat1w = (const float*)din[base + 1];
  a.cat2b = (const float*)din[base + 2];  a.cat2w = (const float*)din[base + 3];
  a.bnbe  = (const float*)din[base + 4];  a.bng   = (const float*)din[base + 5];
  a.kb    = (const float*)din[base + 6];  a.kw    = (const float*)din[base + 7];
  a.mhb   = (const float*)din[base + 8];  a.mhw   = (const float*)din[base + 9];
  a.qb    = (const float*)din[base + 10]; a.qw    = (const float*)din[base + 11];
  a.vb    = (const float*)din[base + 12]; a.vw    = (const float*)din[base + 13];
  return a;
}

struct WSP {
  float *wq, *wk, *wv, *watt, *wadd, *wcat, *wt;
  float *stm, *sts;
};

static void run_attn(const AP& p, const float* d1, int d1bs, const float* d2, int d2bs,
                     int P1, int P2, float* outp, int obs, const WSP& w, hipStream_t st) {
  dim3 blk(32);
  // q, k, v projections
  k_conv_wmma<<<dim3(P1 / 16, Cc / 16, Bb), blk, 0, st>>>(
      d1, p.qw, p.qb, (const float*)nullptr, w.wq, Cc, Cc, P1, d1bs, Cc * P1, 0, Cc);
  k_conv_wmma<<<dim3(P2 / 16, Cc / 16, Bb), blk, 0, st>>>(
      d2, p.kw, p.kb, (const float*)nullptr, w.wk, Cc, Cc, P2, d2bs, Cc * P2, 0, Cc);
  k_conv_wmma<<<dim3(P2 / 16, Cc / 16, Bb), blk, 0, st>>>(
      d2, p.vw, p.vb, (const float*)nullptr, w.wv, Cc, Cc, P2, d2bs, Cc * P2, 0, Cc);
  // scores + softmax + apply
  k_attn_score<<<dim3(P2 / 16, P1 / 16, Bb * Hh), blk, 0, st>>>(
      w.wq, w.wk, w.watt, P1, P2, 0.1767766953f);
  k_softmax<<<dim3(Bb * Hh * P1), dim3(256), 0, st>>>(w.watt, P2);
  k_attn_apply<<<dim3(P1 / 16, Hh * 2, Bb), blk, 0, st>>>(w.watt, w.wv, w.wadd, P1, P2);
  // mh projection into upper half of cat; copy d1 into lower half
  k_conv_wmma<<<dim3(P1 / 16, Cc / 16, Bb), blk, 0, st>>>(
      w.wadd, p.mhw, p.mhb, (const float*)nullptr, w.wcat + (size_t)Cc * P1,
      Cc, Cc, P1, Cc * P1, C2 * P1, 0, Cc);
  int tot = Bb * Cc * P1;
  k_copy_chan<<<dim3((tot + 255) / 256), dim3(256), 0, st>>>(d1, d1bs, w.wcat, C2 * P1, Cc, P1);
  // cat1 conv + BN + ReLU
  k_conv_wmma<<<dim3(P1 / 16, C2 / 16, Bb), blk, 0, st>>>(
      w.wcat, p.cat1w, p.cat1b, (const float*)nullptr, w.wt, C2, C2, P1,
      C2 * P1, C2 * P1, 0, C2);
  k_bnstats<<<dim3(C2), dim3(256), 0, st>>>(w.wt, P1, C2 * P1, w.stm, w.sts, 1e-5f);
  int tot2 = Bb * C2 * P1;
  k_bnrelu<<<dim3((tot2 + 255) / 256), dim3(256), 0, st>>>(w.wt, C2, P1, C2 * P1,
                                                           w.stm, w.sts, p.bng, p.bnbe);
  // cat2 conv + residual
  k_conv_wmma<<<dim3(P1 / 16, Cc / 16, Bb), blk, 0, st>>>(
      w.wt, p.cat2w, p.cat2b, d1, outp, Cc, C2, P1, C2 * P1, obs, d1bs, C2);
}

struct LearnWS {
  float *lq, *lk, *qn, *kn, *Aq, *Ak, *Atq, *Atk, *Sq, *Sk;
  float *m1, *v1, *m2, *v2, *vbar, *s2c;
};

static void run_learn(const float* d1, const float* d2, float* swout,
                      const float* lqw, const float* lqb, const float* lkw, const float* lkb,
                      const float* bn1g, const float* bn1be, const float* bn2g, const float* bn2be,
                      const float* c1w, const float* c1b, const float* c2w, const float* c2b,
                      const float* shw, const float* shb, const LearnWS& w, hipStream_t st) {
  dim3 blk(32);
  // q/k projections (B,C,S)
  k_conv_wmma<<<dim3(Ss / 16, Cc / 16, Bb), blk, 0, st>>>(
      d1, lqw, lqb, (const float*)nullptr, w.lq, Cc, Cc, Ss, Cc * Ss, Cc * Ss, 0, Cc);
  k_conv_wmma<<<dim3(Ss / 16, Cc / 16, Bb), blk, 0, st>>>(
      d2, lkw, lkb, (const float*)nullptr, w.lk, Cc, Cc, Ss, Cc * Ss, Cc * Ss, 0, Cc);
  // stage-1 IN+BN+ReLU on separable factors
  k_mv_bc<<<dim3(Bb * Cc), dim3(128), 0, st>>>(w.lq, Ss, Cc * Ss, Cc, w.m1, w.v1);
  k_vbar<<<dim3(1), dim3(128), 0, st>>>(w.v1, w.vbar, Cc, 1e-3f);
  int t1 = Bb * Cc * Ss;
  k_in_bn_relu<<<dim3((t1 + 255) / 256), dim3(256), 0, st>>>(
      w.lq, w.qn, w.m1, w.v1, w.vbar, bn1g, bn1be, Cc, Ss, 0);
  k_mv_bc<<<dim3(Bb * Cc), dim3(128), 0, st>>>(w.lk, Ss, Cc * Ss, Cc, w.m2, w.v2);
  k_vbar<<<dim3(1), dim3(128), 0, st>>>(w.v2, w.vbar, Cc, 1e-3f);
  k_in_bn_relu<<<dim3((t1 + 255) / 256), dim3(256), 0, st>>>(
      w.lk, w.kn, w.m2, w.v2, w.vbar, bn1g, bn1be, Cc, Ss, Cc);
  // c1 conv split over the two channel groups (separable): Aq + Ak
  k_conv_wmma<<<dim3(Ss / 16, C2 / 16, Bb), blk, 0, st>>>(
      w.qn, c1w, c1b, (const float*)nullptr, w.Aq, C2, Cc, Ss, Cc * Ss, C2 * Ss, 0, C2);
  k_conv_wmma<<<dim3(Ss / 16, C2 / 16, Bb), blk, 0, st>>>(
      w.kn, c1w + Cc, (const float*)nullptr, (const float*)nullptr, w.Ak,
      C2, Cc, Ss, Cc * Ss, C2 * Ss, 0, C2);
  // stage-2 IN+BN closed-form stats and apply
  k_mv_bc<<<dim3(Bb * C2), dim3(128), 0, st>>>(w.Aq, Ss, C2 * Ss, C2, w.m1, w.v1);
  k_mv_bc<<<dim3(Bb * C2), dim3(128), 0, st>>>(w.Ak, Ss, C2 * Ss, C2, w.m2, w.v2);
  k_bn2scale<<<dim3(1), dim3(256), 0, st>>>(w.v1, w.v2, bn2g, w.s2c);
  int t2 = Bb * C2 * Ss;
  k_stage2_apply<<<dim3((t2 + 255) / 256), dim3(256), 0, st>>>(
      w.Aq, w.m1, w.v1, w.v2, w.s2c, bn2be, w.Atq);
  k_stage2_apply<<<dim3((t2 + 255) / 256), dim3(256), 0, st>>>(
      w.Ak, w.m2, w.v1, w.v2, w.s2c, (const float*)nullptr, w.Atk);
  // shortcut dots on raw q/k (plus all scalar biases into Sq)
  int td = Bb * Ss;
  k_dot1<<<dim3((td + 255) / 256), dim3(256), 0, st>>>(w.lq, shw, shb, c2b, w.Sq, Cc, Ss, Cc * Ss);
  k_dot1<<<dim3((td + 255) / 256), dim3(256), 0, st>>>(w.lk, shw + Cc, (const float*)nullptr,
                                                       (const float*)nullptr, w.Sk, Cc, Ss, Cc * Ss);
  // fused rank-reduce over channels + softmax over m
  k_sepsum<<<dim3(Ss, Bb), dim3(128), 0, st>>>(w.Atq, w.Atk, c2w, w.Sq, w.Sk, swout);
  k_softmax<<<dim3(Bb * Ss), dim3(256), 0, st>>>(swout, Ss);
}

extern "C" void kernel_launch(void* const* d_in, const int* in_sizes, int n_in,
                              void* d_out, int out_size, void* d_ws, size_t ws_size,
                              hipStream_t stream) {
  (void)in_sizes; (void)n_in; (void)out_size;
  const float* desc1 = (const float*)d_in[0];
  const float* desc2 = (const float*)d_in[1];
  const int* si1 = (const int*)d_in[2];
  const int* si2 = (const int*)d_in[3];
  // params flattened in jax dict-sorted order
  const float* clf1b = (const float*)d_in[4];  const float* clf1w = (const float*)d_in[5];
  const float* clf2b = (const float*)d_in[6];  const float* clf2w = (const float*)d_in[7];
  const float* clfbe = (const float*)d_in[8];  const float* clfg  = (const float*)d_in[9];
  AP cross = ap_from(d_in, 10);
  AP down  = ap_from(d_in, 24);
  const float* lkb = (const float*)d_in[38]; const float* lkw = (const float*)d_in[39];
  const float* lqb = (const float*)d_in[40]; const float* lqw = (const float*)d_in[41];
  const float* bn1be = (const float*)d_in[42]; const float* bn1g = (const float*)d_in[43];
  const float* bn2be = (const float*)d_in[44]; const float* bn2g = (const float*)d_in[45];
  const float* c1b = (const float*)d_in[46]; const float* c1w = (const float*)d_in[47];
  const float* c2b = (const float*)d_in[48]; const float* c2w = (const float*)d_in[49];
  const float* shb = (const float*)d_in[50]; const float* shw = (const float*)d_in[51];
  AP selfp = ap_from(d_in, 52);
  AP up    = ap_from(d_in, 66);

  float* out = (float*)d_out;
  float* ws = (float*)d_ws;

  const size_t NBC = (size_t)Bb * Cc * Nn;   // 2,048,000
  const size_t SBC = (size_t)Bb * Cc * Ss;   // 131,072
  const size_t SB2C = 2 * SBC;               // 262,144
  const size_t ATT = (size_t)Bb * Hh * Nn * Ss; // 8,192,000

  size_t off = 0;
  auto grab = [&](size_t n) { float* p = ws + off; off += (n + 15) & ~(size_t)15; return p; };
  float* wq   = grab(NBC);
  float* wk   = grab(NBC);
  float* wv   = grab(NBC);
  float* watt = grab(ATT);           // also reused as cat(first half) + t(second half)
  float* wcat = watt;
  float* wt   = watt + (size_t)Bb * C2 * Nn;
  float* wadd = wq;                  // q is dead after score kernel
  float* cl1  = grab(SBC);
  float* cl2  = grab(SBC);
  float* cc   = grab(SB2C);
  float* cc2  = grab(SB2C);
  float* ccx  = grab(SB2C);
  float* s1   = grab(SBC);
  float* s2b  = grab(SBC);
  LearnWS lw;
  lw.lq = grab(SBC); lw.lk = grab(SBC); lw.qn = grab(SBC); lw.kn = grab(SBC);
  lw.Aq = grab(SB2C); lw.Ak = grab(SB2C); lw.Atq = grab(SB2C); lw.Atk = grab(SB2C);
  lw.Sq = grab(Bb * Ss); lw.Sk = grab(Bb * Ss);
  float* stm = grab(C2); float* sts = grab(C2);
  lw.m1 = grab(Bb * C2); lw.v1 = grab(Bb * C2);
  lw.m2 = grab(Bb * C2); lw.v2 = grab(Bb * C2);
  lw.vbar = grab(C2); lw.s2c = grab(C2);
  if (ws_size < off * sizeof(float)) return;  // workspace too small

  WSP w{wq, wk, wv, watt, wadd, wcat, wt, stm, sts};
  dim3 blk(32);

  // 1) gather seeds
  int tg = Bb * Cc * Ss;
  k_gather<<<dim3((tg + 255) / 256), dim3(256), 0, stream>>>(desc1, si1, cl1);
  k_gather<<<dim3((tg + 255) / 256), dim3(256), 0, stream>>>(desc2, si2, cl2);

  // 2) down attention -> halves of cc
  run_attn(down, cl1, Cc * Ss, desc1, Cc * Nn, Ss, Nn, cc, C2 * Ss, w, stream);
  run_attn(down, cl2, Cc * Ss, desc2, Cc * Nn, Ss, Nn, cc + (size_t)Cc * Ss, C2 * Ss, w, stream);

  // 3) classifier MLP: cc2 = clf2( relu( bn( clf1(cc) ) ) )
  k_conv_wmma<<<dim3(Ss / 16, C2 / 16, Bb), blk, 0, stream>>>(
      cc, clf1w, clf1b, (const float*)nullptr, wt, C2, C2, Ss, C2 * Ss, C2 * Ss, 0, C2);
  k_bnstats<<<dim3(C2), dim3(256), 0, stream>>>(wt, Ss, C2 * Ss, stm, sts, 1e-5f);
  int tc = Bb * C2 * Ss;
  k_bnrelu<<<dim3((tc + 255) / 256), dim3(256), 0, stream>>>(wt, C2, Ss, C2 * Ss,
                                                             stm, sts, clfg, clfbe);
  k_conv_wmma<<<dim3(Ss / 16, C2 / 16, Bb), blk, 0, stream>>>(
      wt, clf2w, clf2b, (const float*)nullptr, cc2, C2, C2, Ss, C2 * Ss, C2 * Ss, 0, C2);

  // 4) cross attention -> halves of ccx
  run_attn(cross, cc2, C2 * Ss, cc2 + (size_t)Cc * Ss, C2 * Ss, Ss, Ss, ccx, C2 * Ss, w, stream);
  run_attn(cross, cc2 + (size_t)Cc * Ss, C2 * Ss, cc2, C2 * Ss, Ss, Ss,
           ccx + (size_t)Cc * Ss, C2 * Ss, w, stream);

  // 5) self attention -> s1, s2
  run_attn(selfp, ccx, C2 * Ss, ccx, C2 * Ss, Ss, Ss, s1, Cc * Ss, w, stream);
  run_attn(selfp, ccx + (size_t)Cc * Ss, C2 * Ss, ccx + (size_t)Cc * Ss, C2 * Ss,
           Ss, Ss, s2b, Cc * Ss, w, stream);

  // 6) learn_score -> sw1, sw2 (written directly into output, softmaxed in place)
  float* swout = out + 2 * NBC;
  run_learn(s1, s2b, swout, lqw, lqb, lkw, lkb, bn1g, bn1be, bn2g, bn2be,
            c1w, c1b, c2w, c2b, shw, shb, lw, stream);
  run_learn(s2b, s1, swout + (size_t)Bb * Ss * Ss, lqw, lqb, lkw, lkb, bn1g, bn1be,
            bn2g, bn2be, c1w, c1b, c2w, c2b, shw, shb, lw, stream);

  // 7) up attention -> d1n, d2n
  run_attn(up, desc1, Cc * Nn, s1, Cc * Ss, Nn, Ss, out, Cc * Nn, w, stream);
  run_attn(up, desc2, Cc * Nn, s2b, Cc * Ss, Nn, Ss, out + NBC, Cc * Nn, w, stream);
}